// QbcStroke_24635932410160
// MI455X (gfx1250) — compile-verified
//
#include <hip/hip_runtime.h>

#define WIDTH 512
#define NSTEP 100
#define NPAD  112              // 7 K-tiles of 16 steps
#define NTK   (NPAD/16)        // 7 step tiles
#define TILE  16               // output tile edge
#define HALO  18               // tile + 1-pixel blur halo
#define HPIX  (HALO*HALO)      // 324 halo pixels
#define NGRP  ((HPIX + 15)/16) // 21 pixel groups of 16

typedef __attribute__((ext_vector_type(2))) float v2f;
typedef __attribute__((ext_vector_type(8))) float v8f;

// Evaluate stroke at parameter index tIdx: center (X,Y), radius r, color c
__device__ __forceinline__ void evalStep(const float* __restrict__ s, int tIdx,
                                         float& X, float& Y, float& r, float& c)
{
    float p0x = s[0], p0y = s[1];
    float p1x = s[2], p1y = s[3];
    float p2x = s[4], p2y = s[5];
    float q1x = p0x + (p2x - p0x) * p1x;   // P1 = P0 + (P2-P0)*P1
    float q1y = p0y + (p2y - p0y) * p1y;
    const float W1 = (float)(WIDTH - 1);
    float X0 = p0x * W1 + 0.5f, Y0 = p0y * W1 + 0.5f;
    float X1 = q1x * W1 + 0.5f, Y1 = q1y * W1 + 0.5f;
    float X2 = p2x * W1 + 0.5f, Y2 = p2y * W1 + 0.5f;
    float r0  = 1.0f + s[6] * (float)(WIDTH / 4);
    float r2e = 1.0f + s[7] * (float)(WIDTH / 4);

    float t  = (float)tIdx * (1.0f / (float)(NSTEP - 1)); // linspace(0,1,100)
    float mt = 1.0f - t;
    float qs = mt * mt, qm = 2.0f * mt * t, qe = t * t;
    X = qs * X0 + qm * X1 + qe * X2;
    Y = qs * Y0 + qm * Y1 + qe * Y2;
    r = r0 * mt + r2e * t;
    c = s[8] * mt + s[9] * t;
}

__global__ __launch_bounds__(256) void qbc_render(const float* __restrict__ strokes,
                                                  float* __restrict__ out)
{
    // A-matrix rows, both K-halves concatenated:
    //   sA[t]        = {2X, 2Y}              (K=0,1)
    //   sA[NPAD + t] = {-1, r^2 - X^2 - Y^2} (K=2,3)
    __shared__ v2f    sA[2 * NPAD];
    __shared__ float  sC[NPAD + 1];   // step colors + sentinel 0 at [NPAD]
    __shared__ float4 sBB[8];         // per-tile center bbox {minX,maxX,minY,maxY}
    __shared__ float  sBBr[8];        // per-tile max radius
    __shared__ float  sCv[HPIX];      // canvas halo tile

    const int tid = threadIdx.x;
    const int b  = blockIdx.z;
    const int i0 = blockIdx.x * TILE;
    const int j0 = blockIdx.y * TILE;
    const float* s = strokes + b * 10;

    // ---- Phase A: step tables + per-tile culling aggregates -------------
    if (tid < NPAD) {
        if (tid < NSTEP) {
            float X, Y, r, c;
            evalStep(s, tid, X, Y, r, c);
            v2f lo; lo.x = 2.0f * X;  lo.y = 2.0f * Y;
            v2f hi; hi.x = -1.0f;     hi.y = r * r - X * X - Y * Y;
            sA[tid]        = lo;
            sA[NPAD + tid] = hi;
            sC[tid] = c;
        } else {
            v2f lo; lo.x = 0.0f;  lo.y = 0.0f;
            v2f hi; hi.x = -1.0f; hi.y = -1.0e30f;   // pad: can never hit
            sA[tid]        = lo;
            sA[NPAD + tid] = hi;
            sC[tid] = 0.0f;
        }
    } else if (tid < NPAD + NTK) {
        // per-step-tile conservative bounds (real steps only)
        const int tk = tid - NPAD;
        float mnX = 1.0e30f, mxX = -1.0e30f, mnY = 1.0e30f, mxY = -1.0e30f, mxR = 0.0f;
        for (int k = 0; k < 16; ++k) {
            int t = tk * 16 + k;
            if (t < NSTEP) {
                float X, Y, r, c;
                evalStep(s, t, X, Y, r, c);
                mnX = fminf(mnX, X); mxX = fmaxf(mxX, X);
                mnY = fminf(mnY, Y); mxY = fmaxf(mxY, Y);
                mxR = fmaxf(mxR, r);
            }
        }
        sBB[tk]  = make_float4(mnX, mxX, mnY, mxY);
        sBBr[tk] = mxR;
    } else if (tid == NPAD + NTK) {
        sC[NPAD] = 0.0f;                 // sentinel color for "no hit"
    }
    __syncthreads();

    // ---- Phase B: WMMA rasterization of the 18x18 halo ------------------
    const int  lane  = tid & 31;
    const int  wave  = __builtin_amdgcn_readfirstlane(tid >> 5);  // scalar loop var
    const bool upper = (lane >= 16);          // K-half / M-half selector (wave32)
    const int  m     = lane & 15;
    const int  aOff  = (lane & 16) * (NPAD / 16) + m;  // upper -> +NPAD
    const int  bi    = (lane < NTK) ? lane : 0;        // tile tested by this lane

    // Loop-invariant LDS data: hoist out of the group loop.
    // 7 A fragments (14 VGPRs) + this lane's tile bbox for culling.
    v2f a0 = sA[aOff +   0], a1 = sA[aOff +  16], a2 = sA[aOff +  32];
    v2f a3 = sA[aOff +  48], a4 = sA[aOff +  64], a5 = sA[aOff +  80];
    v2f a6 = sA[aOff +  96];
    const float4 bb = sBB[bi];
    const float  rr = sBBr[bi];

    for (int g = wave; g < NGRP; g += 8) {    // scalar bound: EXEC stays all-ones
        int p  = g * 16 + m;            // flat halo pixel = this lane's B column
        int hi = p / HALO;
        int hj = p - hi * HALO;
        int ci = i0 + hi - 1;           // canvas coords (may be OOB -> zero pad)
        int cj = j0 + hj - 1;
        float fi = (float)ci, fj = (float)cj;

        // B (4x16): lanes 0-15 rows {i, j}; lanes 16-31 rows {i^2+j^2, 1}
        v2f bv;
        bv.x = upper ? (fi * fi + fj * fj) : fi;
        bv.y = upper ? 1.0f                : fj;

        // --- conservative tile-skip mask (wave-uniform) -----------------
        int pLo  = g * 16;
        int hiLo = pLo / HALO, hiHi = (pLo + 15) / HALO;
        float piMin = (float)(i0 + hiLo - 1);
        float piMax = (float)(i0 + hiHi - 1);
        bool  oneRow = (hiLo == hiHi);
        float pjMin = (float)(j0 - 1 + (oneRow ? (pLo - hiLo * HALO) : 0));
        float pjMax = oneRow ? (pjMin + 15.0f) : (float)(j0 + TILE);
        float dx = fmaxf(0.0f, fmaxf(bb.x - piMax, piMin - bb.y));
        float dy = fmaxf(0.0f, fmaxf(bb.z - pjMax, pjMin - bb.w));
        bool  canHit = (dx * dx + dy * dy <= rr * rr) && (lane < NTK);
        unsigned mm = __builtin_amdgcn_ballot_w32(canHit);
        int   smask = __builtin_amdgcn_readfirstlane((int)mm);

        // 8 independent max-t accumulators -> short dependency chains
        float b0 = -1.0f, b1 = -1.0f, b2 = -1.0f, b3 = -1.0f;
        float b4 = -1.0f, b5 = -1.0f, b6 = -1.0f, b7 = -1.0f;

        #pragma unroll
        for (int tk = 0; tk < NTK; ++tk) {
            if (smask & (1 << tk)) {        // scalar branch, EXEC untouched
                v2f av = (tk == 0) ? a0 : (tk == 1) ? a1 : (tk == 2) ? a2 :
                         (tk == 3) ? a3 : (tk == 4) ? a4 : (tk == 5) ? a5 : a6;
                v8f cz = {};
                // d[v] = r^2 - dist^2 for step (tk*16 + v + 8*upper), pixel (lane&15)
                v8f d = __builtin_amdgcn_wmma_f32_16x16x4_f32(
                    false, av, false, bv, (short)0, cz, false, false);
                const int tOff = tk * 16 + (upper ? 8 : 0);
                b0 = (d[0] > 0.0f) ? (float)(tOff + 0) : b0;
                b1 = (d[1] > 0.0f) ? (float)(tOff + 1) : b1;
                b2 = (d[2] > 0.0f) ? (float)(tOff + 2) : b2;
                b3 = (d[3] > 0.0f) ? (float)(tOff + 3) : b3;
                b4 = (d[4] > 0.0f) ? (float)(tOff + 4) : b4;
                b5 = (d[5] > 0.0f) ? (float)(tOff + 5) : b5;
                b6 = (d[6] > 0.0f) ? (float)(tOff + 6) : b6;
                b7 = (d[7] > 0.0f) ? (float)(tOff + 7) : b7;
            }
        }
        // max-tree merge (last hit == largest t), then cross-half merge
        float bestT = fmaxf(fmaxf(fmaxf(b0, b1), fmaxf(b2, b3)),
                            fmaxf(fmaxf(b4, b5), fmaxf(b6, b7)));
        bestT = fmaxf(bestT, __shfl_xor(bestT, 16, 32));

        int   ti  = (bestT >= 0.0f) ? (int)bestT : NPAD;  // sentinel -> color 0
        float val = sC[ti];
        if (ci < 0 || ci >= WIDTH || cj < 0 || cj >= WIDTH) val = 0.0f; // zero pad
        if (!upper && p < HPIX) sCv[p] = val;
    }
    __syncthreads();

    // ---- Phase C: fused 3x3 Gaussian blur + (1 - canvas) ----------------
    int oi = tid >> 4, oj = tid & 15;          // 256 threads = 16x16 outputs
    const float e2  = 0.13533528323661270f;    // exp(-2)
    const float s1  = 1.0f + 2.0f * e2;
    const float inv = 1.0f / (s1 * s1);        // separable kernel normalizer
    const float w1[3] = {e2, 1.0f, e2};
    float acc = 0.0f;
    #pragma unroll
    for (int di = 0; di < 3; ++di)
        #pragma unroll
        for (int dj = 0; dj < 3; ++dj)
            acc += w1[di] * w1[dj] * sCv[(oi + di) * HALO + (oj + dj)];
    acc *= inv;

    int i = i0 + oi, j = j0 + oj;
    out[((size_t)b * WIDTH + i) * WIDTH + j] = 1.0f - acc;
}

extern "C" void kernel_launch(void* const* d_in, const int* in_sizes, int n_in,
                              void* d_out, int out_size, void* d_ws, size_t ws_size,
                              hipStream_t stream) {
    (void)in_sizes; (void)n_in; (void)d_ws; (void)ws_size; (void)out_size;
    const float* strokes = (const float*)d_in[0];
    float* out = (float*)d_out;
    dim3 grid(WIDTH / TILE, WIDTH / TILE, 128);
    qbc_render<<<grid, 256, 0, stream>>>(strokes, out);
}